// PeriodicConvolutionWithKernel_33758442947104
// MI455X (gfx1250) — compile-verified
//
#include <hip/hip_runtime.h>
#include <hip/hip_bf16.h>

typedef float v2f __attribute__((ext_vector_type(2)));
typedef float v8f __attribute__((ext_vector_type(8)));

#define N_NODES_  10000
#define N_EDGES_  250000
#define N_RBF_    10
#define FEAT_IN_  144
#define Y_DIM_    25
#define P_TOTAL_  4864
#define NPATHS_   19
#define TILE_E_   16
#define THREADS_  256

// PATHS table: (io, ii, lf) in reference order; do=2*io+1, di=2*ii+1, df=2*lf+1
__constant__ int c_io[NPATHS_]    = {0,0,0,1,1,1,1,1,1,1,2,2,2,2,2,2,2,2,2};
__constant__ int c_ii[NPATHS_]    = {0,1,2,0,1,1,1,2,2,2,0,1,1,1,2,2,2,2,2};
__constant__ int c_lf[NPATHS_]    = {0,1,2,1,0,1,2,1,2,3,2,1,2,3,0,1,2,3,4};
__constant__ int c_cgoff[NPATHS_] = {0,1,10,35,44,53,80,125,170,245,350,375,420,495,600,625,700,825,1000};
__constant__ int c_foff[3] = {0,16,64};   // FEAT_OFF per l_in
__constant__ int c_cb[3]   = {0,1,4};     // packed output-column base per l_out
// writeout maps for packed columns 0..8 -> (block offset, do, o_within_block)
__constant__ int c_outb[9] = {0,16,16,16,64,64,64,64,64};
__constant__ int c_odo[9]  = {1,3,3,3,5,5,5,5,5};
__constant__ int c_orel[9] = {0,0,1,2,0,1,2,3,4};

__global__ __launch_bounds__(THREADS_)
void se3_tp_conv_kernel(const float* __restrict__ features,
                        const float* __restrict__ R,
                        const float* __restrict__ Ys,
                        const float* __restrict__ radii,
                        const float* __restrict__ cg_flat,
                        const float* __restrict__ n_norm,
                        const int*   __restrict__ map_a,
                        const int*   __restrict__ map_b,
                        float*       __restrict__ out)
{
    __shared__ __align__(16) float F_lds[TILE_E_ * FEAT_IN_];   // 2304 gathered source feats
    __shared__ __align__(16) float Ys_lds[TILE_E_ * Y_DIM_];    // 400
    __shared__ __align__(16) float radii_lds[TILE_E_ * 12];     // 192 (K padded 10->12 w/ zeros)
    __shared__ __align__(16) float cg_lds[240];                 // current path CG block (<=225)
    __shared__ __align__(16) float cgY_lds[TILE_E_ * 25];       // per-edge cgY[o*di+i]
    __shared__ __align__(16) float tmp_lds[TILE_E_ * 256];      // tmp_t[e][o][v] (B operand)
    __shared__ __align__(16) float W_lds[TILE_E_ * 256];        // W[e][w*16+v]   (A operand)

    const int tid   = threadIdx.x;
    const int lane  = tid & 31;
    const int wid   = tid >> 5;
    const int lm    = lane & 15;
    const int khi   = (lane < 16) ? 0 : 2;            // K half-select for A/B operands
    const int erow  = (lane < 16) ? 0 : 8;            // D row offset
    const int ebase = blockIdx.x * TILE_E_;

    // ---------------- stage tile into LDS ----------------
    for (int idx = tid; idx < TILE_E_ * FEAT_IN_; idx += THREADS_) {
        int e = idx / FEAT_IN_, c = idx - e * FEAT_IN_;
        int eg = ebase + e; if (eg >= N_EDGES_) eg = N_EDGES_ - 1;
        F_lds[idx] = features[(size_t)map_b[eg] * FEAT_IN_ + c];
    }
    for (int idx = tid; idx < TILE_E_ * Y_DIM_; idx += THREADS_) {
        int e = idx / Y_DIM_, c = idx - e * Y_DIM_;
        int eg = ebase + e; if (eg >= N_EDGES_) eg = N_EDGES_ - 1;
        Ys_lds[idx] = Ys[(size_t)eg * Y_DIM_ + c];
    }
    if (tid < TILE_E_ * 12) {
        int e = tid / 12, k = tid - e * 12;
        int eg = ebase + e; if (eg >= N_EDGES_) eg = N_EDGES_ - 1;
        radii_lds[tid] = (k < N_RBF_) ? radii[(size_t)eg * N_RBF_ + k] : 0.0f;
    }
    __syncthreads();

    // persistent per-wave fp32 accumulators: wave owns edges 2*wid and 2*wid+1.
    // D[w, col]: cols 0..8 hold packed (l_out, o) message outputs.
    v8f acc0 = {0,0,0,0,0,0,0,0};
    v8f acc1 = {0,0,0,0,0,0,0,0};

    for (int p = 0; p < NPATHS_; ++p) {
        const int   io   = c_io[p], ii = c_ii[p], lf = c_lf[p];
        const int   dO   = 2 * io + 1, di = 2 * ii + 1, df = 2 * lf + 1;
        const int   foff = c_foff[ii];
        const int   cb   = c_cb[io];
        const float norm = rsqrtf((float)df);

        if (p + 1 < NPATHS_)  // warm L2/WGP$ for next path's R block
            __builtin_prefetch(&R[(size_t)(p + 1) * 256], 0, 1);

        // phase 1: CG block -> LDS
        const int ncg = dO * di * df;
        if (tid < ncg) cg_lds[tid] = cg_flat[c_cgoff[p] + tid];
        __syncthreads();

        // phase 2: cgY[e][o*di+i] = norm * sum_f cg[o,i,f] * Y[e, lf^2+f]
        const int pd = dO * di;
        for (int idx = tid; idx < TILE_E_ * pd; idx += THREADS_) {
            int e = idx / pd, j = idx - e * pd;
            const float* yp  = &Ys_lds[e * Y_DIM_ + lf * lf];
            const float* cgp = &cg_lds[j * df];
            float s = 0.f;
            for (int f = 0; f < df; ++f) s += cgp[f] * yp[f];
            cgY_lds[e * 25 + j] = norm * s;
        }
        __syncthreads();

        // phase 3: tmp_t[e][o][v] = sum_i F[e, foff+v*di+i] * cgY[e, (o-cb)*di+i]
        // zero outside the path's packed column window [cb, cb+dO)
        {
            const int o = tid >> 4, v = tid & 15;
            const int orel = o - cb;
            const bool act = (orel >= 0) && (orel < dO);
            for (int e = 0; e < TILE_E_; ++e) {
                float s = 0.f;
                if (act) {
                    const float* fp = &F_lds[e * FEAT_IN_ + foff + v * di];
                    const float* cp = &cgY_lds[e * 25 + orel * di];
                    for (int i = 0; i < di; ++i) s += fp[i] * cp[i];
                }
                tmp_lds[e * 256 + tid] = s;   // tid == o*16 + v
            }
        }

        // phase 4: W-GEMM via WMMA: W[e][m] = sum_k radii[e][k] * R[k][p*256+m]
        // M = 16 edges, N = 256 cols (16 tiles, 2 per wave), K = 12 (rows 10,11 zero in A)
        {
            for (int tt = wid; tt < 16; tt += 8) {
                const int c0 = tt * 16;
                v8f d = {0,0,0,0,0,0,0,0};
                #pragma unroll
                for (int kb = 0; kb < 3; ++kb) {
                    v2f a = *(const v2f*)&radii_lds[lm * 12 + kb * 4 + khi];
                    const int row = kb * 4 + khi;
                    v2f b;
                    if (row < N_RBF_) {
                        const float* rp = &R[(size_t)row * P_TOTAL_ + p * 256 + c0 + lm];
                        b.x = rp[0];
                        b.y = rp[P_TOTAL_];
                    } else { b.x = 0.f; b.y = 0.f; }
                    d = __builtin_amdgcn_wmma_f32_16x16x4_f32(false, a, false, b,
                                                              (short)0, d, false, false);
                }
                #pragma unroll
                for (int j = 0; j < 8; ++j)
                    W_lds[(size_t)(j + erow) * 256 + c0 + lm] = d[j];
            }
        }
        __syncthreads();

        // phase 5: per-edge message WMMA: msg[w, col] += sum_v W[e][w,v] * tmp_t[e][col,v]
        // A lanes = M rows(w), B lanes = N cols(col); K = v, 4 steps of K=4
        {
            const int off = lm * 16 + khi;    // even float index -> 8B aligned b64 loads
            const int e0 = 2 * wid, e1 = e0 + 1;
            #pragma unroll
            for (int vb = 0; vb < 16; vb += 4) {
                v2f a = *(const v2f*)&W_lds[e0 * 256 + off + vb];
                v2f b = *(const v2f*)&tmp_lds[e0 * 256 + off + vb];
                acc0 = __builtin_amdgcn_wmma_f32_16x16x4_f32(false, a, false, b,
                                                             (short)0, acc0, false, false);
            }
            #pragma unroll
            for (int vb = 0; vb < 16; vb += 4) {
                v2f a = *(const v2f*)&W_lds[e1 * 256 + off + vb];
                v2f b = *(const v2f*)&tmp_lds[e1 * 256 + off + vb];
                acc1 = __builtin_amdgcn_wmma_f32_16x16x4_f32(false, a, false, b,
                                                             (short)0, acc1, false, false);
            }
        }
        __syncthreads();
    }

    // ---------------- writeout: scale by n_norm[dst] and scatter-add ----------------
    for (int q = 0; q < 2; ++q) {
        const int e  = 2 * wid + q;
        const int eg = ebase + e;
        if (eg < N_EDGES_ && lm < 9) {
            const int   a  = map_a[eg];
            const float nn = n_norm[a];
            const v8f acc = q ? acc1 : acc0;
            const int ob = c_outb[lm], od = c_odo[lm], orr = c_orel[lm];
            #pragma unroll
            for (int j = 0; j < 8; ++j) {
                const int w = j + erow;
                atomicAdd(&out[(size_t)a * FEAT_IN_ + ob + w * od + orr], acc[j] * nn);
            }
        }
    }
}

extern "C" void kernel_launch(void* const* d_in, const int* in_sizes, int n_in,
                              void* d_out, int out_size, void* d_ws, size_t ws_size,
                              hipStream_t stream) {
    (void)in_sizes; (void)n_in; (void)d_ws; (void)ws_size;
    const float* features = (const float*)d_in[0];
    const float* R        = (const float*)d_in[1];
    const float* Ys       = (const float*)d_in[2];
    const float* radii    = (const float*)d_in[3];
    const float* cg_flat  = (const float*)d_in[4];
    const float* n_norm   = (const float*)d_in[5];
    const int*   map_a    = (const int*)d_in[6];
    const int*   map_b    = (const int*)d_in[7];
    float* out = (float*)d_out;

    hipMemsetAsync(out, 0, (size_t)out_size * sizeof(float), stream);

    const int ntiles = (N_EDGES_ + TILE_E_ - 1) / TILE_E_;   // 15625
    se3_tp_conv_kernel<<<ntiles, THREADS_, 0, stream>>>(
        features, R, Ys, radii, cg_flat, n_norm, map_a, map_b, out);
}